// FunctionModel_206158430579
// MI455X (gfx1250) — compile-verified
//
#include <hip/hip_runtime.h>
#include <hip/hip_bf16.h>
#include <math.h>

#define N_POINTS 8192
#define BATCH 64
#define INPUT_DIM (2 * N_POINTS)
#define N_TILES (N_POINTS / 16)   // 512
#define BIG_F 100000000.0f
#define EPS_F 1e-08f

typedef __attribute__((ext_vector_type(2))) float v2f;
typedef __attribute__((ext_vector_type(8))) float v8f;

// ---------------------------------------------------------------------------
// Kernel 1: ku_x[i] = clip(clip(q0,±10) * 2cos(2x_i), ±1e6), same for y.
// Only coords row 0 feeds the NN-derivative path (reference uses ku[0]).
// ---------------------------------------------------------------------------
__global__ __launch_bounds__(256) void precompute_kernel(
    const float* __restrict__ coords, const float* __restrict__ xs,
    const float* __restrict__ ys, float* __restrict__ kux,
    float* __restrict__ kuy) {
  int i = blockIdx.x * 256 + threadIdx.x;
  if (i >= N_POINTS) return;
  float q = coords[i];                       // row 0, first N_POINTS entries
  q = fminf(10.0f, fmaxf(-10.0f, q));
  float ux = 2.0f * __cosf(2.0f * xs[i]);
  float uy = 2.0f * __cosf(2.0f * ys[i]);
  float kx = q * ux, ky = q * uy;
  kux[i] = fminf(1.0e6f, fmaxf(-1.0e6f, kx));
  kuy[i] = fminf(1.0e6f, fmaxf(-1.0e6f, ky));
}

// ---------------------------------------------------------------------------
// Kernel 2: 1-D nearest-neighbor + divided difference via WMMA, both
// coordinates in ONE launch (blocks 0..63 -> x, 64..127 -> y) so 1024 waves
// are resident at once.
// (x_i - x_j)^2 = [x^2, x, 1, 0] . [1, -2x', x'^2, 0]^T  -> one
// V_WMMA_F32_16X16X4_F32 per 16x16 distance tile. One wave per 16-row
// i-tile (itile forced into an SGPR via readfirstlane so the tile loops are
// scalar-branch only); diagonal tile handled once before the loops.
// ---------------------------------------------------------------------------
__global__ __launch_bounds__(256) void nn_deriv_kernel(
    const float* __restrict__ xg0, const float* __restrict__ ku0,
    float* __restrict__ dku0,       // coordinate 0 (x)
    const float* __restrict__ xg1, const float* __restrict__ ku1,
    float* __restrict__ dku1) {     // coordinate 1 (y)
  __shared__ float lds_x[N_POINTS + 16];   // +16: prefetch over-read pad

  // scalar coordinate select: blocks [0,64) -> x, [64,128) -> y
  const int bid = blockIdx.x;
  const bool isy = bid >= (N_TILES / 8);
  const float* __restrict__ xg  = isy ? xg1  : xg0;
  const float* __restrict__ ku  = isy ? ku1  : ku0;
  float* __restrict__       dku = isy ? dku1 : dku0;
  const int btile = isy ? (bid - N_TILES / 8) : bid;

  const int tid = threadIdx.x;
  // cooperative load of all 8192 coords into LDS (32 KB)
  {
    const float4* src = (const float4*)xg;
    float4* dst = (float4*)lds_x;
    for (int v = tid; v < N_POINTS / 4; v += 256) dst[v] = src[v];
    if (tid < 16) lds_x[N_POINTS + tid] = 0.0f;  // pad
  }
  __syncthreads();

  const int lane = tid & 31;
  const int col  = lane & 15;             // N / M index within tile
  const int hi   = lane >> 4;             // 0: lanes 0-15, 1: lanes 16-31
  // wave-uniform tile id, forced scalar so loop bounds/branches stay on SALU
  const int itile = __builtin_amdgcn_readfirstlane(btile * 8 + (tid >> 5));
  const int ibase = itile * 16;

  // A operand (16x4, row i = ibase+col holds [x^2, x, 1, 0]):
  //   VGPR0 = K0 (lo) / K2 (hi), VGPR1 = K1 (lo) / K3 (hi)
  const float xi = lds_x[ibase + col];
  v2f a;
  a.x = hi ? 1.0f : xi * xi;
  a.y = hi ? 0.0f : xi;

  float best[8];
  int   bidx[8];

  // ---- diagonal tile first: seeds best/bidx, masks self-distance ----
  {
    const float xj = lds_x[ibase + col];
    v2f b;
    b.x = hi ? xj * xj : 1.0f;
    b.y = hi ? 0.0f    : -2.0f * xj;
    v8f c = {};
    v8f d = __builtin_amdgcn_wmma_f32_16x16x4_f32(
        false, a, false, b, (short)0, c, false, false);
    const int jcol = ibase + col;
#pragma unroll
    for (int k = 0; k < 8; ++k) {
      float dk = d[k];
      if (col == k + hi * 8) dk = BIG_F;  // j == i on the diagonal tile
      best[k] = dk;
      bidx[k] = jcol;
    }
  }

  // ---- remaining 511 tiles: branch-free body, pipelined LDS feed ----
#define NN_RANGE(LO, HI)                                                    \
  {                                                                         \
    const int lo_ = (LO), hi_ = (HI);                                       \
    if (lo_ < hi_) {                                                        \
      float xcur = lds_x[lo_ * 16 + col];                                   \
      for (int jt = lo_; jt < hi_; ++jt) {                                  \
        const float xnext = lds_x[(jt + 1) * 16 + col]; /* pad-safe */      \
        v2f b;                                                              \
        b.x = hi ? xcur * xcur : 1.0f;                                      \
        b.y = hi ? 0.0f        : -2.0f * xcur;                              \
        v8f c = {};                                                         \
        v8f d = __builtin_amdgcn_wmma_f32_16x16x4_f32(                      \
            false, a, false, b, (short)0, c, false, false);                 \
        const int jcol = jt * 16 + col;                                     \
        _Pragma("unroll")                                                   \
        for (int k = 0; k < 8; ++k) {                                       \
          const float dk = d[k];                                            \
          if (dk < best[k]) { best[k] = dk; bidx[k] = jcol; }               \
        }                                                                   \
        xcur = xnext;                                                       \
      }                                                                     \
    }                                                                       \
  }

  NN_RANGE(0, itile)
  NN_RANGE(itile + 1, N_TILES)
#undef NN_RANGE

  // Cross-lane argmin within each 16-lane half (ties -> lowest index,
  // matching jnp.argmin), then the divided difference for this row.
#pragma unroll
  for (int k = 0; k < 8; ++k) {
    float bd = best[k];
    int   bi = bidx[k];
#pragma unroll
    for (int m = 1; m < 16; m <<= 1) {
      const float od = __shfl_xor(bd, m, 32);
      const int   oi = __shfl_xor(bi, m, 32);
      if (od < bd || (od == bd && oi < bi)) { bd = od; bi = oi; }
    }
    if (col == 0) {
      const int i = ibase + k + hi * 8;
      const float dg = ku[i] - ku[bi];
      const float dc = xg[i] - xg[bi];
      dku[i] = dg / (dc + EPS_F);
    }
  }
}

// ---------------------------------------------------------------------------
// Kernel 3a: U = 0.5 * sum((f_obs - clip(dkx+dky, ±200))^2)  (scalar)
// ---------------------------------------------------------------------------
__global__ __launch_bounds__(256) void u_reduce_kernel(
    const float* __restrict__ fobs, const float* __restrict__ dkx,
    const float* __restrict__ dky, float* __restrict__ Uout) {
  __shared__ float sbuf[256];
  float acc = 0.0f;
  for (int i = threadIdx.x; i < N_POINTS; i += 256) {
    float fh = dkx[i] + dky[i];
    fh = fminf(200.0f, fmaxf(-200.0f, fh));
    const float diff = fobs[i] - fh;
    acc += diff * diff;
  }
  sbuf[threadIdx.x] = acc;
  __syncthreads();
  for (int s = 128; s > 0; s >>= 1) {
    if (threadIdx.x < s) sbuf[threadIdx.x] += sbuf[threadIdx.x + s];
    __syncthreads();
  }
  if (threadIdx.x == 0) Uout[0] = 0.5f * sbuf[0];
}

// ---------------------------------------------------------------------------
// Kernel 3b: out[b] = U + 0.5 * sum_i p[b,i]^2   (one block per batch row)
// ---------------------------------------------------------------------------
__global__ __launch_bounds__(256) void out_kernel(
    const float* __restrict__ coords, const float* __restrict__ Uin,
    float* __restrict__ out) {
  const int b = blockIdx.x;
  const float* p = coords + (size_t)b * INPUT_DIM + N_POINTS;
  __shared__ float sbuf[256];
  float acc = 0.0f;
  for (int i = threadIdx.x; i < N_POINTS; i += 256) {
    const float v = p[i];
    acc += v * v;
  }
  sbuf[threadIdx.x] = acc;
  __syncthreads();
  for (int s = 128; s > 0; s >>= 1) {
    if (threadIdx.x < s) sbuf[threadIdx.x] += sbuf[threadIdx.x + s];
    __syncthreads();
  }
  if (threadIdx.x == 0) out[b] = Uin[0] + 0.5f * sbuf[0];
}

// ---------------------------------------------------------------------------
extern "C" void kernel_launch(void* const* d_in, const int* in_sizes, int n_in,
                              void* d_out, int out_size, void* d_ws,
                              size_t ws_size, hipStream_t stream) {
  const float* coords = (const float*)d_in[0];  // (64, 16384)
  const float* xs     = (const float*)d_in[1];  // (1, 8192)
  const float* ys     = (const float*)d_in[2];  // (1, 8192)
  const float* fobs   = (const float*)d_in[3];  // (1, 8192)
  float* out = (float*)d_out;                   // (64, 1)

  float* ws  = (float*)d_ws;
  float* kux = ws;                  // 8192
  float* kuy = ws + N_POINTS;      // 8192
  float* dkx = ws + 2 * N_POINTS;  // 8192
  float* dky = ws + 3 * N_POINTS;  // 8192
  float* U   = ws + 4 * N_POINTS;  // 1

  precompute_kernel<<<N_POINTS / 256, 256, 0, stream>>>(coords, xs, ys, kux, kuy);
  // both coordinates in one launch: blocks [0,64) -> x, [64,128) -> y
  nn_deriv_kernel<<<2 * (N_TILES / 8), 256, 0, stream>>>(xs, kux, dkx,
                                                         ys, kuy, dky);
  u_reduce_kernel<<<1, 256, 0, stream>>>(fobs, dkx, dky, U);
  out_kernel<<<BATCH, 256, 0, stream>>>(coords, U, out);
}